// DLGeneEmbeddings_5480378270213
// MI455X (gfx1250) — compile-verified
//
#include <hip/hip_runtime.h>

typedef __attribute__((ext_vector_type(2))) float v2f;
typedef __attribute__((ext_vector_type(4))) float v4f;
typedef __attribute__((ext_vector_type(8))) float v8f;

#define BATCH        1024
#define NUM_GENES    20000
#define NONGENE_DIM  64
#define EMB_DIM      32
#define INPUT_DIM    (NUM_GENES + NONGENE_DIM)   // 20064

// ---------------------------------------------------------------------------
// GEMV via V_WMMA_F32_16X16X4_F32 (wave32):
//   out[r] = dot(A[r, col_off : col_off+K], w[0:K]) + bias
// One wave computes 16 rows. A-tile layout (16x4 f32):
//   lanes 0-15 : M=lane,    VGPR0=K0, VGPR1=K1
//   lanes 16-31: M=lane-16, VGPR0=K2, VGPR1=K3
// B (4x16) holds the weight chunk replicated in every column, so every column
// of D is the same dot product. D layout: lanes 0-15 VGPRv -> row v,
// lanes 16-31 VGPRv -> row v+8.
// ---------------------------------------------------------------------------
__global__ void gemv_wmma_f32(const float* __restrict__ A, long long row_stride,
                              long long col_off, int n_rows, int K,
                              const float* __restrict__ w,
                              const float* __restrict__ bias_ptr,
                              float* __restrict__ out) {
  const int wave = (int)(blockIdx.x * (blockDim.x >> 5) + (threadIdx.x >> 5));
  const int lane = (int)(threadIdx.x & 31);
  const int n_tiles = n_rows >> 4;           // n_rows is a multiple of 16
  if (wave >= n_tiles) return;               // wave-uniform exit: EXEC stays all-1s

  const int row   = (wave << 4) + (lane & 15);
  const int kpair = (lane >> 4) << 1;        // 0 for lanes 0-15, 2 for lanes 16-31
  const float* arow = A + (long long)row * row_stride + col_off + kpair;
  const float* wrow = w + kpair;

  v8f c = {};
  for (int k0 = 0; k0 < K; k0 += 4) {
    v2f a, bv;
    a.x  = arow[k0];  a.y  = arow[k0 + 1];   // A[row][k0+kpair], A[row][k0+kpair+1]
    bv.x = wrow[k0];  bv.y = wrow[k0 + 1];   // w replicated across the 16 columns
    // 8 args: (neg_a, A, neg_b, B, c_mod, C, reuse_a, reuse_b)
    c = __builtin_amdgcn_wmma_f32_16x16x4_f32(false, a, false, bv, (short)0, c,
                                              false, false);
  }

  const float bias = bias_ptr ? bias_ptr[0] : 0.0f;
  if (lane == 0) {
#pragma unroll
    for (int v = 0; v < 8; ++v) out[(wave << 4) + v] = c[v] + bias;
  } else if (lane == 16) {
#pragma unroll
    for (int v = 0; v < 8; ++v) out[(wave << 4) + 8 + v] = c[v] + bias;
  }
}

// ---------------------------------------------------------------------------
// Fused bandwidth kernel: y[i,g] = x[i,g]*w_x + (gp[g] + ng[i])
// x / y are streamed once -> non-temporal 128-bit accesses (keep L2 for gp).
// ---------------------------------------------------------------------------
__global__ void fused_stream_kernel(const float* __restrict__ x,
                                    const float* __restrict__ gp,   // gene_term + b
                                    const float* __restrict__ ng,   // ng_term
                                    const float* __restrict__ W,
                                    float* __restrict__ out) {
  const int col4 = blockIdx.x * blockDim.x + threadIdx.x;    // float4 column index
  if (col4 >= NUM_GENES / 4) return;
  const int row = blockIdx.y;

  const float w_x = W[NONGENE_DIM];        // uniform -> scalar load
  const float ngr = ng[row];               // uniform per block

  const long long xoff = (long long)row * INPUT_DIM + (long long)col4 * 4;
  const long long ooff = (long long)row * NUM_GENES + (long long)col4 * 4;

  v4f xv = __builtin_nontemporal_load((const v4f*)(x + xoff));   // NT: no reuse
  v4f gv = *(const v4f*)(gp + (long long)col4 * 4);              // RT: L2-resident
  v4f y  = xv * w_x + (gv + ngr);
  __builtin_nontemporal_store(y, (v4f*)(out + ooff));            // NT: no reuse
}

// ---------------------------------------------------------------------------
extern "C" void kernel_launch(void* const* d_in, const int* in_sizes, int n_in,
                              void* d_out, int out_size, void* d_ws, size_t ws_size,
                              hipStream_t stream) {
  const float* x   = (const float*)d_in[0];   // (1024, 20064)
  const float* emb = (const float*)d_in[1];   // (20000, 32)
  const float* W   = (const float*)d_in[2];   // (97,)
  const float* b   = (const float*)d_in[3];   // scalar (1 elem)
  float* out = (float*)d_out;

  // Workspace layout: gp[NUM_GENES] then ng[BATCH]  (~84 KB)
  float* gp = (float*)d_ws;
  float* ng = gp + NUM_GENES;

  // 1) gp[g] = dot(emb[g, :], W[65:97]) + b      (20000/16 = 1250 wave-tiles)
  {
    const int tiles = NUM_GENES / 16;
    const int wpb = 8;                                  // 8 waves / 256 threads
    dim3 grd((tiles + wpb - 1) / wpb), blk(32 * wpb);
    gemv_wmma_f32<<<grd, blk, 0, stream>>>(emb, EMB_DIM, 0, NUM_GENES, EMB_DIM,
                                           W + NONGENE_DIM + 1, b, gp);
  }

  // 2) ng[i] = dot(x[i, 20000:20064], W[0:64])   (1024/16 = 64 wave-tiles)
  {
    const int tiles = BATCH / 16;
    const int wpb = 8;
    dim3 grd((tiles + wpb - 1) / wpb), blk(32 * wpb);
    gemv_wmma_f32<<<grd, blk, 0, stream>>>(x, INPUT_DIM, NUM_GENES, BATCH,
                                           NONGENE_DIM, W, nullptr, ng);
  }

  // 3) y = x*w_x + gp + ng  (HBM-bound: 160 MB streamed, ~7 us at 23.3 TB/s)
  {
    dim3 blk(256);
    dim3 grd((NUM_GENES / 4 + 255) / 256, BATCH);       // (20, 1024)
    fused_stream_kernel<<<grd, blk, 0, stream>>>(x, gp, ng, W, out);
  }
}